// Net_317827579949
// MI455X (gfx1250) — compile-verified
//
#include <hip/hip_runtime.h>
#include <hip/hip_bf16.h>
#include <math.h>

#define Nn 100000
#define Ee 1600000
#define Bb 256
#define NEGV (-1e30f)

typedef float v2f __attribute__((ext_vector_type(2)));
typedef float v8f __attribute__((ext_vector_type(8)));

// ---- order-preserving float <-> uint encoding for atomic max ----
__device__ __forceinline__ unsigned encf(float f) {
  unsigned u = __float_as_uint(f);
  return (u & 0x80000000u) ? ~u : (u | 0x80000000u);
}
__device__ __forceinline__ float decf(unsigned u) {
  return (u & 0x80000000u) ? __uint_as_float(u & 0x7FFFFFFFu) : __uint_as_float(~u);
}

// ---- fills ----
__global__ void fill_f32(float* __restrict__ p, float v, long long n) {
  long long i = blockIdx.x * 256LL + threadIdx.x;
  if (i < n) p[i] = v;
}
__global__ void fill_enc(unsigned* __restrict__ p, float v, int n) {
  int i = blockIdx.x * 256 + threadIdx.x;
  if (i < n) p[i] = encf(v);
}

// ---- edge scatter-add: agg[dst] += x[src] * emask (emask==nullptr -> 1) ----
template<int F>
__global__ void scatter_add_k(const float* __restrict__ xin, const int* __restrict__ src,
                              const int* __restrict__ dst, const float* __restrict__ emask,
                              float* __restrict__ agg) {
  const long long total = (long long)Ee * (F / 4);
  long long id = blockIdx.x * 256LL + threadIdx.x;
  if (id >= total) return;
  const int CH = F / 4;
  int e = (int)(id / CH);
  int c = (int)(id - (long long)e * CH);
  float em = emask ? emask[e] : 1.0f;
  if (em == 0.0f) return;
  const float4 v = *(const float4*)(xin + (size_t)src[e] * F + c * 4);
  float* o = agg + (size_t)dst[e] * F + c * 4;
  atomicAdd(o + 0, v.x * em);
  atomicAdd(o + 1, v.y * em);
  atomicAdd(o + 2, v.z * em);
  atomicAdd(o + 3, v.w * em);
}

__global__ void add_inplace_k(float* __restrict__ d, const float* __restrict__ s, long long n) {
  long long i = blockIdx.x * 256LL + threadIdx.x;
  if (i < n) d[i] += s[i];
}

// ---- WMMA fp32 GEMM: out[M,64] = relu(A[M,K] @ W[K,64] + bias), M = Mtiles*16 ----
// LDS image of W is interleaved by k-pairs so each lane's B fragment is a
// contiguous v2f: sW[((r/2)*64 + c)*2 + (r&1)] = W[r][c]. Since (k+koff) is
// always even, lane reads ds_load_b64 at ((k+koff)/2*64 + col)*8 bytes, with
// the 4 column tiles at +32/+64/+96 float immediate offsets.
template<int K>
__global__ __launch_bounds__(256)
void gemm_bias_relu(const float* __restrict__ A, const float* __restrict__ W,
                    const float* __restrict__ bias, float* __restrict__ out, int Mtiles) {
  __shared__ float sW[K * 64];
  for (int i = threadIdx.x; i < K * 64; i += 256) {
    int r = i >> 6, c = i & 63;
    sW[(((r >> 1) << 6) + c) * 2 + (r & 1)] = W[i];
  }
  __syncthreads();
  const int wave = threadIdx.x >> 5;
  const int lane = threadIdx.x & 31;
  const int tile = blockIdx.x * 8 + wave;
  if (tile >= Mtiles) return;           // wave-uniform: EXEC stays all-ones for WMMA
  const int hi = lane >> 4;             // 0: K=0,1 / M rows 0..7 ; 1: K=2,3 / M rows 8..15
  const int l  = lane & 15;
  const int row  = tile * 16 + l;
  const int koff = hi * 2;
  v8f acc0 = {0,0,0,0,0,0,0,0};
  v8f acc1 = acc0, acc2 = acc0, acc3 = acc0;
  const float* arow = A + (size_t)row * K + koff;
  for (int k = 0; k < K; k += 4) {
    v2f a = *(const v2f*)(arow + k);                 // A[row][k+koff .. k+koff+1]
    const int kk = (k + koff) >> 1;                  // k-pair index
    const float* bp = sW + (((kk << 6) + l) << 1);   // col tile 0, lane col l
    v2f b0 = *(const v2f*)(bp);
    v2f b1 = *(const v2f*)(bp + 32);
    v2f b2 = *(const v2f*)(bp + 64);
    v2f b3 = *(const v2f*)(bp + 96);
    acc0 = __builtin_amdgcn_wmma_f32_16x16x4_f32(false, a, false, b0, (short)0, acc0, false, false);
    acc1 = __builtin_amdgcn_wmma_f32_16x16x4_f32(false, a, false, b1, (short)0, acc1, false, false);
    acc2 = __builtin_amdgcn_wmma_f32_16x16x4_f32(false, a, false, b2, (short)0, acc2, false, false);
    acc3 = __builtin_amdgcn_wmma_f32_16x16x4_f32(false, a, false, b3, (short)0, acc3, false, false);
  }
  const float bb0 = bias[l], bb1 = bias[l + 16], bb2 = bias[l + 32], bb3 = bias[l + 48];
  #pragma unroll
  for (int r = 0; r < 8; ++r) {
    const int m = hi * 8 + r;
    float* orow = out + (size_t)(tile * 16 + m) * 64 + l;
    orow[0]  = fmaxf(acc0[r] + bb0, 0.0f);
    orow[16] = fmaxf(acc1[r] + bb1, 0.0f);
    orow[32] = fmaxf(acc2[r] + bb2, 0.0f);
    orow[48] = fmaxf(acc3[r] + bb3, 0.0f);
  }
}

// ---- per-node dot with 64-vector: hg[i] = h[i,:] . w ----
__global__ void node_dot64_k(const float* __restrict__ h, const float* __restrict__ w,
                             float* __restrict__ outv, int n) {
  int i = blockIdx.x * 256 + threadIdx.x;
  if (i >= n) return;
  const float4* hp = (const float4*)(h + (size_t)i * 64);
  const float4* wp = (const float4*)w;
  float acc = 0.f;
  #pragma unroll
  for (int j = 0; j < 16; ++j) {
    float4 a = hp[j], b = wp[j];
    acc += a.x * b.x + a.y * b.y + a.z * b.z + a.w * b.w;
  }
  outv[i] = acc;
}

// ---- GCN attention pieces ----
__global__ void deg_init_k(float* __restrict__ deg, const float* __restrict__ nmask, int n) {
  int i = blockIdx.x * 256 + threadIdx.x;
  if (i < n) deg[i] = nmask ? nmask[i] : 1.0f;
}
__global__ void deg_edges_k(float* __restrict__ deg, const int* __restrict__ dst,
                            const float* __restrict__ em, int ne) {
  int e = blockIdx.x * 256 + threadIdx.x;
  if (e >= ne) return;
  float m = em[e];
  if (m != 0.f) atomicAdd(&deg[dst[e]], m);
}
__global__ void attn_init_k(float* __restrict__ attn, float* __restrict__ dinv,
                            const float* __restrict__ deg, const float* __restrict__ hg,
                            const float* __restrict__ nmask, const float* __restrict__ gb, int n) {
  int i = blockIdx.x * 256 + threadIdx.x;
  if (i >= n) return;
  float d = deg[i];
  float di = (d > 0.f) ? rsqrtf(fmaxf(d, 1e-12f)) : 0.f;
  dinv[i] = di;
  float nm = nmask ? nmask[i] : 1.0f;
  attn[i] = hg[i] * di * di * nm + gb[0];
}
__global__ void attn_edges_k(float* __restrict__ attn, const float* __restrict__ hg,
                             const float* __restrict__ dinv, const int* __restrict__ src,
                             const int* __restrict__ dst, const float* __restrict__ em, int ne) {
  int e = blockIdx.x * 256 + threadIdx.x;
  if (e >= ne) return;
  float m = em[e];
  if (m == 0.f) return;
  float c = dinv[src[e]] * dinv[dst[e]] * m;
  if (c != 0.f) atomicAdd(&attn[dst[e]], hg[src[e]] * c);
}

// ---- segment softmax + top-k keep ----
__global__ void segmax_k(const float* __restrict__ attn, const float* __restrict__ nmask,
                         const int* __restrict__ batch, unsigned* __restrict__ mEnc, int n) {
  int i = blockIdx.x * 256 + threadIdx.x;
  if (i >= n) return;
  float nm = nmask ? nmask[i] : 1.f;
  atomicMax(&mEnc[batch[i]], encf(nm > 0.f ? attn[i] : NEGV));
}
__global__ void expsum_k(const float* __restrict__ attn, const float* __restrict__ nmask,
                         const int* __restrict__ batch, const unsigned* __restrict__ mEnc,
                         float* __restrict__ ev, float* __restrict__ sSum, int n) {
  int i = blockIdx.x * 256 + threadIdx.x;
  if (i >= n) return;
  float nm = nmask ? nmask[i] : 1.f;
  float e = 0.f;
  if (nm > 0.f) e = expf(attn[i] - decf(mEnc[batch[i]]));
  ev[i] = e;
  if (e != 0.f) atomicAdd(&sSum[batch[i]], e);
}
__global__ void score_k(const float* __restrict__ ev, const float* __restrict__ sSum,
                        const int* __restrict__ batch, const float* __restrict__ nmask,
                        float* __restrict__ score, unsigned* __restrict__ smaxEnc, int n) {
  int i = blockIdx.x * 256 + threadIdx.x;
  if (i >= n) return;
  float s = ev[i] / fmaxf(sSum[batch[i]], 1e-30f);
  score[i] = s;
  float nm = nmask ? nmask[i] : 1.f;
  atomicMax(&smaxEnc[batch[i]], encf(nm > 0.f ? s : NEGV));
}
__global__ void keep_kk(const float* __restrict__ score, const unsigned* __restrict__ smaxEnc,
                        const float* __restrict__ nmask, const int* __restrict__ batch,
                        float* __restrict__ keep, int n) {
  int i = blockIdx.x * 256 + threadIdx.x;
  if (i >= n) return;
  float thr = fminf(1e-3f, decf(smaxEnc[batch[i]]) - 1e-7f);
  float nm = nmask ? nmask[i] : 1.f;
  keep[i] = (nm > 0.f && score[i] > thr) ? 1.f : 0.f;
}
__global__ void scale_x2_k(const float* __restrict__ h, const float* __restrict__ score,
                           const float* __restrict__ keep, float* __restrict__ o, long long n64) {
  long long id = blockIdx.x * 256LL + threadIdx.x;
  if (id >= n64) return;
  int i = (int)(id >> 6);
  o[id] = h[id] * (score[i] * keep[i]);
}
__global__ void mask_edges_k(float* __restrict__ em, const float* __restrict__ keep,
                             const int* __restrict__ src, const int* __restrict__ dst, int ne) {
  int e = blockIdx.x * 256 + threadIdx.x;
  if (e >= ne) return;
  em[e] *= keep[src[e]] * keep[dst[e]];
}

// ---- global max pool + final linear ----
__global__ void pool_max_k(const float* __restrict__ h, const float* __restrict__ keep2,
                           const int* __restrict__ batch, unsigned* __restrict__ gEnc, long long n64) {
  long long id = blockIdx.x * 256LL + threadIdx.x;
  if (id >= n64) return;
  int i = (int)(id >> 6);
  int f = (int)(id & 63);
  if (keep2[i] > 0.f) atomicMax(&gEnc[(size_t)batch[i] * 64 + f], encf(h[id]));
}
__global__ void out_linear_k(const unsigned* __restrict__ gEnc, const float* __restrict__ Wl,
                             const float* __restrict__ bl, float* __restrict__ outp, int nb) {
  int b = blockIdx.x * 256 + threadIdx.x;
  if (b >= nb) return;
  float acc = bl[0];
  const unsigned* g = gEnc + (size_t)b * 64;
  #pragma unroll
  for (int f = 0; f < 64; ++f) acc += decf(g[f]) * Wl[f];
  outp[b] = acc;
}

// ---- single-block inclusive scan: r1 = clip(cumsum(keep1)-1, 0, n-1) ----
__global__ void scan_r1_kernel(const float* __restrict__ keep, int* __restrict__ r1, int n) {
  __shared__ float tile[256];
  __shared__ float carry;
  if (threadIdx.x == 0) carry = 0.0f;
  __syncthreads();
  for (int base = 0; base < n; base += 256) {
    int i = base + (int)threadIdx.x;
    tile[threadIdx.x] = (i < n) ? keep[i] : 0.0f;
    __syncthreads();
    for (int off = 1; off < 256; off <<= 1) {
      float add = (threadIdx.x >= (unsigned)off) ? tile[threadIdx.x - off] : 0.0f;
      __syncthreads();
      tile[threadIdx.x] += add;
      __syncthreads();
    }
    float inc = tile[threadIdx.x] + carry;
    if (i < n) {
      int idx = (int)inc - 1;
      idx = idx < 0 ? 0 : (idx > n - 1 ? n - 1 : idx);
      r1[i] = idx;
    }
    __syncthreads();
    if (threadIdx.x == 0) carry += tile[255];
    __syncthreads();
  }
}

// ---- attention KL loss accumulation ----
__global__ void kl_k(const float* __restrict__ na, const int* __restrict__ r1,
                     const float* __restrict__ score2, const float* __restrict__ keep2,
                     const int* __restrict__ batch, float* __restrict__ sB,
                     float* __restrict__ cB, int n) {
  int i = blockIdx.x * 256 + threadIdx.x;
  if (i >= n) return;
  float k2 = keep2[i];
  float t = na[r1[i]];
  float logp = logf(score2[i] + 1e-14f);
  float kl = ((t > 0.f) ? t * logf(fmaxf(t, 1e-38f)) : 0.f) - t * logp;
  kl *= k2;
  if (kl != 0.f) atomicAdd(&sB[batch[i]], kl);
  if (k2 != 0.f) atomicAdd(&cB[batch[i]], k2);
}
__global__ void finalize_k(const float* __restrict__ sB, const float* __restrict__ cB,
                           float* __restrict__ outp) {
  __shared__ float red[256];
  int b = threadIdx.x;                      // one block of 256 == Bb
  float c = cB[b];
  outp[256 + b] = sB[b] / fmaxf(c, 1.0f);   // attn_loss
  red[b] = c;
  __syncthreads();
  for (int off = 128; off > 0; off >>= 1) {
    if (b < off) red[b] += red[b + off];
    __syncthreads();
  }
  if (b == 0) outp[512] = red[0] / (float)Nn;   // ratio
}

extern "C" void kernel_launch(void* const* d_in, const int* in_sizes, int n_in,
                              void* d_out, int out_size, void* d_ws, size_t ws_size,
                              hipStream_t stream) {
  (void)in_sizes; (void)n_in; (void)out_size; (void)ws_size;
  const float* x     = (const float*)d_in[0];
  const int*   src   = (const int*)d_in[1];
  const int*   dstp  = (const int*)d_in[2];
  const int*   batch = (const int*)d_in[3];
  const float* natt  = (const float*)d_in[4];
  const float* W11 = (const float*)d_in[5];  const float* b11 = (const float*)d_in[6];
  const float* W12 = (const float*)d_in[7];  const float* b12 = (const float*)d_in[8];
  const float* gW1 = (const float*)d_in[9];  const float* gb1 = (const float*)d_in[10];
  const float* W21 = (const float*)d_in[11]; const float* b21 = (const float*)d_in[12];
  const float* W22 = (const float*)d_in[13]; const float* b22 = (const float*)d_in[14];
  const float* gW2 = (const float*)d_in[15]; const float* gb2 = (const float*)d_in[16];
  const float* W31 = (const float*)d_in[17]; const float* b31 = (const float*)d_in[18];
  const float* W32 = (const float*)d_in[19]; const float* b32 = (const float*)d_in[20];
  const float* Wl  = (const float*)d_in[21]; const float* bl  = (const float*)d_in[22];
  float* out = (float*)d_out;

  // ---- workspace layout (floats) ----
  float* w = (float*)d_ws;
  size_t o = 0;
  float* X128 = w + o; o += (size_t)Nn * 128;   // h0 for GIN1; aliased as F3 later
  float* F1   = w + o; o += (size_t)Nn * 64;
  float* F2   = w + o; o += (size_t)Nn * 64;
  float* F3   = X128;                            // reuse: X128 dead after GEMM1
  float* emask = w + o; o += Ee;
  float* hg    = w + o; o += Nn;
  float* deg   = w + o; o += Nn;
  float* dinv  = w + o; o += Nn;
  float* attn  = w + o; o += Nn;
  float* ev    = w + o; o += Nn;
  float* score1 = w + o; o += Nn;
  float* keep1  = w + o; o += Nn;
  float* score2 = w + o; o += Nn;
  float* keep2  = w + o; o += Nn;
  int*   r1     = (int*)(w + o); o += Nn;
  unsigned* mEnc    = (unsigned*)(w + o); o += Bb;
  float*    sSum    = w + o; o += Bb;
  unsigned* smaxEnc = (unsigned*)(w + o); o += Bb;
  float*    sB      = w + o; o += Bb;
  float*    cB      = w + o; o += Bb;
  unsigned* gEnc    = (unsigned*)(w + o); o += (size_t)Bb * 64;

  const int TB = 256;
  auto gridFor = [](long long n) { return (unsigned)((n + 255) / 256); };
  const int Mtiles = Nn / 16;                 // 6250, exact
  const unsigned gGemm = (unsigned)((Mtiles + 7) / 8);

  auto run_pool = [&](const float* h, const float* gWp, const float* gbp,
                      const float* nmask, float* score, float* keep, float* x2) {
    node_dot64_k<<<gridFor(Nn), TB, 0, stream>>>(h, gWp, hg, Nn);
    deg_init_k<<<gridFor(Nn), TB, 0, stream>>>(deg, nmask, Nn);
    deg_edges_k<<<gridFor(Ee), TB, 0, stream>>>(deg, dstp, emask, Ee);
    attn_init_k<<<gridFor(Nn), TB, 0, stream>>>(attn, dinv, deg, hg, nmask, gbp, Nn);
    attn_edges_k<<<gridFor(Ee), TB, 0, stream>>>(attn, hg, dinv, src, dstp, emask, Ee);
    fill_enc<<<1, TB, 0, stream>>>(mEnc, NEGV, Bb);
    fill_f32<<<1, TB, 0, stream>>>(sSum, 0.f, Bb);
    segmax_k<<<gridFor(Nn), TB, 0, stream>>>(attn, nmask, batch, mEnc, Nn);
    expsum_k<<<gridFor(Nn), TB, 0, stream>>>(attn, nmask, batch, mEnc, ev, sSum, Nn);
    fill_enc<<<1, TB, 0, stream>>>(smaxEnc, NEGV, Bb);
    score_k<<<gridFor(Nn), TB, 0, stream>>>(ev, sSum, batch, nmask, score, smaxEnc, Nn);
    keep_kk<<<gridFor(Nn), TB, 0, stream>>>(score, smaxEnc, nmask, batch, keep, Nn);
    scale_x2_k<<<gridFor((long long)Nn * 64), TB, 0, stream>>>(h, score, keep, x2, (long long)Nn * 64);
    mask_edges_k<<<gridFor(Ee), TB, 0, stream>>>(emask, keep, src, dstp, Ee);
  };

  // ---- init ----
  fill_f32<<<gridFor(Ee), TB, 0, stream>>>(emask, 1.0f, (long long)Ee);
  fill_f32<<<gridFor((long long)Nn * 128), TB, 0, stream>>>(X128, 0.0f, (long long)Nn * 128);

  // ---- GIN1: h = relu(relu((x+agg)@W11+b11)@W12+b12) ----
  scatter_add_k<128><<<gridFor((long long)Ee * 32), TB, 0, stream>>>(x, src, dstp, nullptr, X128);
  add_inplace_k<<<gridFor((long long)Nn * 128), TB, 0, stream>>>(X128, x, (long long)Nn * 128);
  gemm_bias_relu<128><<<gGemm, TB, 0, stream>>>(X128, W11, b11, F1, Mtiles);
  gemm_bias_relu<64><<<gGemm, TB, 0, stream>>>(F1, W12, b12, F2, Mtiles);

  // ---- SAGPool1 (nmask = ones) -> h = F1 ----
  run_pool(F2, gW1, gb1, nullptr, score1, keep1, F1);

  // ---- GIN2 ----
  fill_f32<<<gridFor((long long)Nn * 64), TB, 0, stream>>>(F2, 0.0f, (long long)Nn * 64);
  scatter_add_k<64><<<gridFor((long long)Ee * 16), TB, 0, stream>>>(F1, src, dstp, emask, F2);
  add_inplace_k<<<gridFor((long long)Nn * 64), TB, 0, stream>>>(F2, F1, (long long)Nn * 64);
  gemm_bias_relu<64><<<gGemm, TB, 0, stream>>>(F2, W21, b21, F3, Mtiles);
  gemm_bias_relu<64><<<gGemm, TB, 0, stream>>>(F3, W22, b22, F2, Mtiles);

  // ---- SAGPool2 (nmask = keep1) -> h = F1 ----
  run_pool(F2, gW2, gb2, keep1, score2, keep2, F1);

  // ---- GIN3 ----
  fill_f32<<<gridFor((long long)Nn * 64), TB, 0, stream>>>(F3, 0.0f, (long long)Nn * 64);
  scatter_add_k<64><<<gridFor((long long)Ee * 16), TB, 0, stream>>>(F1, src, dstp, emask, F3);
  add_inplace_k<<<gridFor((long long)Nn * 64), TB, 0, stream>>>(F3, F1, (long long)Nn * 64);
  gemm_bias_relu<64><<<gGemm, TB, 0, stream>>>(F3, W31, b31, F2, Mtiles);
  gemm_bias_relu<64><<<gGemm, TB, 0, stream>>>(F2, W32, b32, F3, Mtiles);

  // ---- global max pool (masked) + final linear -> out[0..255] ----
  fill_enc<<<gridFor((long long)Bb * 64), TB, 0, stream>>>(gEnc, NEGV, Bb * 64);
  pool_max_k<<<gridFor((long long)Nn * 64), TB, 0, stream>>>(F3, keep2, batch, gEnc, (long long)Nn * 64);
  out_linear_k<<<1, TB, 0, stream>>>(gEnc, Wl, bl, out, Bb);

  // ---- attention KL loss -> out[256..511], ratio -> out[512] ----
  scan_r1_kernel<<<1, TB, 0, stream>>>(keep1, r1, Nn);
  fill_f32<<<1, TB, 0, stream>>>(sB, 0.f, Bb);
  fill_f32<<<1, TB, 0, stream>>>(cB, 0.f, Bb);
  kl_k<<<gridFor(Nn), TB, 0, stream>>>(natt, r1, score2, keep2, batch, sB, cB, Nn);
  finalize_k<<<1, TB, 0, stream>>>(sB, cB, out);
}